// GeoMix1_33440615367380
// MI455X (gfx1250) — compile-verified
//
#include <hip/hip_runtime.h>

typedef float v2f __attribute__((ext_vector_type(2)));
typedef float v8f __attribute__((ext_vector_type(8)));

// ---------------- fills / copies ----------------
__global__ void gm_fill_f(float* __restrict__ p, int n, float v) {
    int i = blockIdx.x * blockDim.x + threadIdx.x;
    if (i < n) p[i] = v;
}
__global__ void gm_fill_i(int* __restrict__ p, int n, int v) {
    int i = blockIdx.x * blockDim.x + threadIdx.x;
    if (i < n) p[i] = v;
}
__global__ void gm_copy_i(const int* __restrict__ a, int* __restrict__ b, int n) {
    int i = blockIdx.x * blockDim.x + threadIdx.x;
    if (i < n) b[i] = a[i];
}

// ---------------- degree on dst (self-loop pre-seeded as 1.0; integer-valued -> exact) ----------------
__global__ void gm_deg(const long long* __restrict__ ei, float* __restrict__ deg, int E) {
    int e = blockIdx.x * blockDim.x + threadIdx.x;
    if (e < E) atomicAdd(&deg[(int)ei[(size_t)E + e]], 1.0f);
}

// ---------------- dinv = deg^-1/2 ; s = (1-a)/deg + a (folds self-loop + alpha residual) ----------------
__global__ void gm_norm(const float* __restrict__ deg, float* __restrict__ dinv,
                        float* __restrict__ s, int N) {
    int i = blockIdx.x * blockDim.x + threadIdx.x;
    if (i < N) {
        float d = deg[i];
        dinv[i] = rsqrtf(d);
        s[i]    = 0.9f / d + 0.1f;
    }
}

// ---------------- CSR build: counts on src ----------------
__global__ void gm_count(const long long* __restrict__ ei, int* __restrict__ cnt, int E) {
    int e = blockIdx.x * blockDim.x + threadIdx.x;
    if (e < E) atomicAdd(&cnt[(int)ei[e]], 1);
}

// ---------------- scan step 1: per-block sums ----------------
__global__ void gm_reduce(const int* __restrict__ cnt, int* __restrict__ partial, int N) {
    __shared__ int sh[256];
    int t = threadIdx.x, i = blockIdx.x * 256 + t;
    sh[t] = (i < N) ? cnt[i] : 0;
    __syncthreads();
    for (int o = 128; o > 0; o >>= 1) {
        if (t < o) sh[t] += sh[t + o];
        __syncthreads();
    }
    if (t == 0) partial[blockIdx.x] = sh[0];
}

// ---------------- scan step 2: serial exclusive scan of the ~391 block partials ----------------
__global__ void gm_scan_partials(int* __restrict__ p, int nb) {
    if (blockIdx.x == 0 && threadIdx.x == 0) {
        int run = 0;
        for (int i = 0; i < nb; i++) { int v = p[i]; p[i] = run; run += v; }
    }
}

// ---------------- scan step 3: in-block exclusive scan + block offset -> rowptr ----------------
__global__ void gm_block_scan(const int* __restrict__ cnt, const int* __restrict__ poff,
                              int* __restrict__ rowptr, int N) {
    __shared__ int sh[256];
    int t = threadIdx.x, i = blockIdx.x * 256 + t;
    int v = (i < N) ? cnt[i] : 0;
    sh[t] = v;
    __syncthreads();
    for (int o = 1; o < 256; o <<= 1) {
        int add = (t >= o) ? sh[t - o] : 0;
        __syncthreads();
        sh[t] += add;
        __syncthreads();
    }
    if (i < N) rowptr[i] = poff[blockIdx.x] + sh[t] - v;   // exclusive
}

// ---------------- CSR fill: col/weight per slot (w_e precomputed once) ----------------
__global__ void gm_csr_fill(const long long* __restrict__ ei, const float* __restrict__ dinv,
                            int* __restrict__ cursor, int* __restrict__ col,
                            float* __restrict__ wv, int E) {
    int e = blockIdx.x * blockDim.x + threadIdx.x;
    if (e < E) {
        int s = (int)ei[e];
        int d = (int)ei[(size_t)E + e];
        int pos = atomicAdd(&cursor[s], 1);
        col[pos] = d;
        wv[pos]  = dinv[s] * dinv[d];
    }
}

// ---------------- CSR gather SpMM, DX=128: one wave per row, lane = float4 chunk ----------------
// row/edge metadata forced wave-uniform (readfirstlane) -> scalar SMEM loads; vector pipe
// does only the 512B row gathers. Plain store, no atomics, no zero-fill.
__global__ void gm_gather_x(const float* __restrict__ cur, float* __restrict__ acc,
                            const int* __restrict__ rowptr, const int* __restrict__ col,
                            const float* __restrict__ wv, int N) {
    int wid = (int)(blockIdx.x * (blockDim.x >> 5) + (threadIdx.x >> 5));
    int row = __builtin_amdgcn_readfirstlane(wid);
    if (row >= N) return;
    int lane = threadIdx.x & 31;
    int e0 = rowptr[row], e1 = rowptr[row + 1];
    float4 a = make_float4(0.f, 0.f, 0.f, 0.f);
    int k = e0;
    for (; k + 1 < e1; k += 2) {
        int   d0 = col[k],     d1 = col[k + 1];
        float w0 = wv[k],      w1 = wv[k + 1];
        const float4 v0 = *(const float4*)(cur + (size_t)d0 * 128 + lane * 4);
        const float4 v1 = *(const float4*)(cur + (size_t)d1 * 128 + lane * 4);
        a.x += w0 * v0.x + w1 * v1.x;
        a.y += w0 * v0.y + w1 * v1.y;
        a.z += w0 * v0.z + w1 * v1.z;
        a.w += w0 * v0.w + w1 * v1.w;
    }
    if (k < e1) {
        int   d0 = col[k];
        float w0 = wv[k];
        const float4 v0 = *(const float4*)(cur + (size_t)d0 * 128 + lane * 4);
        a.x += w0 * v0.x; a.y += w0 * v0.y; a.z += w0 * v0.z; a.w += w0 * v0.w;
    }
    *(float4*)(acc + (size_t)row * 128 + lane * 4) = a;
}

// ---------------- CSR gather SpMM, DY=32: one wave per row, lane = one float ----------------
__global__ void gm_gather_y(const float* __restrict__ cur, float* __restrict__ acc,
                            const int* __restrict__ rowptr, const int* __restrict__ col,
                            const float* __restrict__ wv, int N) {
    int wid = (int)(blockIdx.x * (blockDim.x >> 5) + (threadIdx.x >> 5));
    int row = __builtin_amdgcn_readfirstlane(wid);
    if (row >= N) return;
    int lane = threadIdx.x & 31;
    int e0 = rowptr[row], e1 = rowptr[row + 1];
    float a = 0.f;
    int k = e0;
    for (; k + 1 < e1; k += 2) {
        int   d0 = col[k],   d1 = col[k + 1];
        float w0 = wv[k],    w1 = wv[k + 1];
        a += w0 * cur[(size_t)d0 * 32 + lane] + w1 * cur[(size_t)d1 * 32 + lane];
    }
    if (k < e1) a += wv[k] * cur[(size_t)col[k] * 32 + lane];
    acc[(size_t)row * 32 + lane] = a;
}

// ---------------- WMMA diagonal mix: out_tile = diag(s) * cur_tile + 0.9 * acc_tile ----------------
// Exact fp32 via V_WMMA_F32_16X16X4_F32: four K=4 steps sweep the 16x16 diagonal.
__global__ void gm_wmma_mix(const float* __restrict__ cur, const float* __restrict__ acc,
                            float* __restrict__ out, const float* __restrict__ s,
                            int dim, int tshift, int tiles_total) {
    int wave = blockIdx.x * (blockDim.x >> 5) + (threadIdx.x >> 5);
    if (wave >= tiles_total) return;           // wave-uniform: EXEC stays all-ones
    int lane = threadIdx.x & 31;
    int lm   = lane & 15;
    int half = lane >> 4;
    int m0 = (wave >> tshift) << 4;
    int n0 = (wave & ((1 << tshift) - 1)) << 4;

    v8f c;
#pragma unroll
    for (int r = 0; r < 8; r++)
        c[r] = 0.9f * acc[(size_t)(m0 + r + half * 8) * dim + n0 + lm];

    float sm = s[m0 + lm];
    int k0 = half * 2;
#pragma unroll
    for (int j = 0; j < 4; j++) {
        v2f a, b;
        a.x = (lm == 4 * j + k0)     ? sm : 0.0f;
        a.y = (lm == 4 * j + k0 + 1) ? sm : 0.0f;
        const float* bp = cur + (size_t)(m0 + 4 * j + k0) * dim + n0 + lm;
        b.x = bp[0];
        b.y = bp[dim];
        c = __builtin_amdgcn_wmma_f32_16x16x4_f32(false, a, false, b, (short)0, c, false, false);
    }

#pragma unroll
    for (int r = 0; r < 8; r++)
        out[(size_t)(m0 + r + half * 8) * dim + n0 + lm] = c[r];
}

extern "C" void kernel_launch(void* const* d_in, const int* in_sizes, int n_in,
                              void* d_out, int out_size, void* d_ws, size_t ws_size,
                              hipStream_t stream) {
    const int DX = 128, DY = 32;
    const float*     x0 = (const float*)d_in[0];
    const float*     y0 = (const float*)d_in[1];
    const long long* ei = (const long long*)d_in[2];
    const int N  = in_sizes[0] / DX;
    const int E  = in_sizes[2] / 2;
    const int NB = (N + 255) / 256;            // scan blocks

    float* ox = (float*)d_out;                 // [N*DX]
    float* oy = ox + (size_t)N * DX;           // [N*DY]

    // ---- workspace layout (4-byte units) ----
    float* ws     = (float*)d_ws;
    float* deg    = ws;                        // [N]
    float* dinv   = ws + (size_t)N;            // [N]
    float* sb     = ws + 2 * (size_t)N;        // [N]
    int*   rowptr = (int*)(ws + 3 * (size_t)N);// [N+1]
    int*   cnt    = rowptr + (N + 1);          // [N]
    int*   cursor = cnt + N;                   // [N]
    int*   part   = cursor + N;                // [NB]
    int*   col    = part + NB;                 // [E]
    float* wv     = (float*)(col + E);         // [E]
    size_t off    = (size_t)(3 * (size_t)N) + (N + 1) + N + N + NB + E + E;
    off           = (off + 3) & ~(size_t)3;    // 16B-align the feature buffers
    float* xA     = ws + off;                  // [N*DX]
    float* yA     = xA + (size_t)N * DX;       // [N*DY]

    const int B = 256;

    // ---- normalization terms ----
    gm_fill_f<<<(N + B - 1) / B, B, 0, stream>>>(deg, N, 1.0f);
    gm_deg   <<<(E + B - 1) / B, B, 0, stream>>>(ei, deg, E);
    gm_norm  <<<(N + B - 1) / B, B, 0, stream>>>(deg, dinv, sb, N);

    // ---- CSR build by src: count -> scan -> fill ----
    gm_fill_i<<<(N + B - 1) / B, B, 0, stream>>>(cnt, N, 0);
    gm_count <<<(E + B - 1) / B, B, 0, stream>>>(ei, cnt, E);
    gm_reduce<<<NB, B, 0, stream>>>(cnt, part, N);
    gm_scan_partials<<<1, 32, 0, stream>>>(part, NB);
    gm_block_scan<<<NB, B, 0, stream>>>(cnt, part, rowptr, N);
    gm_fill_i<<<1, 32, 0, stream>>>(rowptr + N, 1, E);
    gm_copy_i<<<(N + B - 1) / B, B, 0, stream>>>(rowptr, cursor, N);
    gm_csr_fill<<<(E + B - 1) / B, B, 0, stream>>>(ei, dinv, cursor, col, wv, E);

    // ---- 4 hops: CSR gather (no atomics) then in-place WMMA diag-mix ----
    const int tilesX = (N / 16) * (DX / 16);
    const int tilesY = (N / 16) * (DY / 16);
    const int gBlocks = (N + 7) / 8;           // 8 waves/block, one wave per row

    const float* curX = x0;
    const float* curY = y0;
    float* pingX[4] = { xA, ox, xA, ox };      // hop 4 lands in d_out
    float* pingY[4] = { yA, oy, yA, oy };

    for (int h = 0; h < 4; h++) {
        float* accX = pingX[h];
        float* accY = pingY[h];

        gm_gather_x<<<gBlocks, B, 0, stream>>>(curX, accX, rowptr, col, wv, N);
        gm_gather_y<<<gBlocks, B, 0, stream>>>(curY, accY, rowptr, col, wv, N);

        gm_wmma_mix<<<(tilesX + 7) / 8, B, 0, stream>>>(curX, accX, accX, sb, DX, 3, tilesX);
        gm_wmma_mix<<<(tilesY + 7) / 8, B, 0, stream>>>(curY, accY, accY, sb, DY, 1, tilesY);

        curX = accX;
        curY = accY;
    }
}